// ICLoss_79809082295156
// MI455X (gfx1250) — compile-verified
//
#include <hip/hip_runtime.h>
#include <math.h>

#define NTOT 8192
#define JSPLIT 2
#define JLEN (NTOT / JSPLIT)      // 4096 floats = 16 KB LDS per block
#define THREADS 256               // 8 waves (wave32)
#define ROWS_PER_BLOCK 128        // 8 waves * 16 rows

typedef __attribute__((ext_vector_type(2))) float v2f;
typedef __attribute__((ext_vector_type(8))) float v8f;
typedef __attribute__((ext_vector_type(4))) int v4i;

typedef __attribute__((address_space(1))) v4i* global_v4i_ptr;
typedef __attribute__((address_space(3))) v4i* lds_v4i_ptr;

#if __has_builtin(__builtin_amdgcn_wmma_f32_16x16x4_f32)
#define HAVE_WMMA_X4 1
#else
#define HAVE_WMMA_X4 0
#endif

#if __has_builtin(__builtin_amdgcn_global_load_async_to_lds_b128)
#define HAVE_ASYNC_LDS 1
#else
#define HAVE_ASYNC_LDS 0
#endif

#if __has_builtin(__builtin_amdgcn_tanhf)
#define HAVE_TANH 1
#else
#define HAVE_TANH 0
#endif

__device__ __forceinline__ float fast_exp2(float x) {
#if __has_builtin(__builtin_amdgcn_exp2f)
  return __builtin_amdgcn_exp2f(x);
#else
  return exp2f(x);
#endif
}

__device__ __forceinline__ float fast_rcp(float x) {
#if __has_builtin(__builtin_amdgcn_rcpf)
  return __builtin_amdgcn_rcpf(x);
#else
  return 1.0f / x;
#endif
}

// si = pre-scaled row value (see row_scale). Returns sigmoid(x_i - x_j).
__device__ __forceinline__ float sigmoid_elem(float si, float xj) {
#if HAVE_TANH
  // sigmoid(d) = 0.5*tanh(d/2) + 0.5 ; si = 0.5*x_i  -> fma, v_tanh, fma
  return __builtin_fmaf(
      0.5f, __builtin_amdgcn_tanhf(__builtin_fmaf(-0.5f, xj, si)), 0.5f);
#else
  // sigmoid(d) = 1/(1+exp2(log2e*(x_j - x_i))) ; si = -log2e*x_i
  return fast_rcp(1.0f +
                  fast_exp2(__builtin_fmaf(1.44269504088896340736f, xj, si)));
#endif
}

__device__ __forceinline__ float row_scale(float xi) {
#if HAVE_TANH
  return 0.5f * xi;
#else
  return -1.44269504088896340736f * xi;
#endif
}

__global__ void __launch_bounds__(THREADS) zero_ws_kernel(float* p, int n) {
  int i = blockIdx.x * blockDim.x + threadIdx.x;
  if (i < n) p[i] = 0.0f;
}

// grid = (NTOT/ROWS_PER_BLOCK, JSPLIT, 2). Accumulates partial soft-rank row
// sums (without the +1 shift, which cancels in the centered correlation) into
// rank[vec*NTOT + row] via atomics.
__global__ void __launch_bounds__(THREADS)
soft_rank_partial(const float* __restrict__ pred,
                  const float* __restrict__ targ,
                  float* __restrict__ rank) {
  __shared__ float hx[JLEN];
  const int tid = threadIdx.x;
  const int vec = blockIdx.z;
  const float* __restrict__ x = vec ? targ : pred;
  float* __restrict__ r = rank + vec * NTOT;
  const int j0 = blockIdx.y * JLEN;

  // ---- Stage this block's j-slice (raw x values) into LDS ----
#if HAVE_ASYNC_LDS
  for (int k = tid; k < JLEN / 4; k += THREADS) {
    __builtin_amdgcn_global_load_async_to_lds_b128(
        (global_v4i_ptr)(x + j0 + k * 4),
        (lds_v4i_ptr)(hx + k * 4),
        /*imm offset*/ 0, /*cpol*/ 0);
  }
#if __has_builtin(__builtin_amdgcn_s_wait_asynccnt)
  __builtin_amdgcn_s_wait_asynccnt(0);
#else
  asm volatile("s_wait_asynccnt 0" ::: "memory");
#endif
#else
  for (int k = tid; k < JLEN; k += THREADS) hx[k] = x[j0 + k];
#endif
  __syncthreads();

  const int wave = tid >> 5;
  const int lane = tid & 31;
  const int rowbase = blockIdx.x * ROWS_PER_BLOCK + wave * 16;
  // A-matrix layout (32-bit 16x4): lanes 0-15 -> M=lane, K=j,j+1 ;
  //                                lanes 16-31 -> M=lane-16, K=j+2,j+3
  const float si = row_scale(x[rowbase + (lane & 15)]);
  const int jsel = (lane >> 4) << 1;  // 0 for low half, 2 for high half

#if HAVE_WMMA_X4
  v8f c0 = {0.f, 0.f, 0.f, 0.f, 0.f, 0.f, 0.f, 0.f};
  v8f c1 = {0.f, 0.f, 0.f, 0.f, 0.f, 0.f, 0.f, 0.f};
  const v2f ones = {1.0f, 1.0f};
#pragma unroll 2
  for (int j = 0; j < JLEN; j += 8) {
    // Broadcast LDS reads: all lanes of each half read the same b64.
    v2f xa = *(const v2f*)(hx + j + jsel);
    v2f xb = *(const v2f*)(hx + j + 4 + jsel);
    v2f a0, a1;
    a0.x = sigmoid_elem(si, xa.x);
    a0.y = sigmoid_elem(si, xa.y);
    a1.x = sigmoid_elem(si, xb.x);
    a1.y = sigmoid_elem(si, xb.y);
    // Row-sum via matrix unit: D = A(16x4) * ones(4x16) + C
    c0 = __builtin_amdgcn_wmma_f32_16x16x4_f32(false, a0, false, ones,
                                               (short)0, c0, false, false);
    c1 = __builtin_amdgcn_wmma_f32_16x16x4_f32(false, a1, false, ones,
                                               (short)0, c1, false, false);
  }
  v8f c = c0 + c1;
  // D layout: VGPR m, lanes 0-15 -> row m (any column); lanes 16-31 -> row m+8.
  // Take column 0 of each half: lane 0 has rows 0..7, lane 16 has rows 8..15.
  if ((lane & 15) == 0) {
    const int half = lane >> 4;
#pragma unroll
    for (int m = 0; m < 8; ++m)
      atomicAdd(&r[rowbase + half * 8 + m], c[m]);
  }
#else
  float acc = 0.0f;
  for (int j = 0; j < JLEN; j += 4) {
    v2f xa = *(const v2f*)(hx + j + jsel);
    acc += sigmoid_elem(si, xa.x);
    acc += sigmoid_elem(si, xa.y);
  }
  // Lanes L and L+16 cover different K for the same row; both accumulate.
  atomicAdd(&r[rowbase + (lane & 15)], acc);
#endif
}

// Single block: centered Pearson correlation of the two rank vectors, negated.
__global__ void __launch_bounds__(THREADS)
ic_finalize(const float* __restrict__ rank, float* __restrict__ out) {
  __shared__ float sA[THREADS];
  __shared__ float sB[THREADS];
  __shared__ float sC[THREADS];
  const int tid = threadIdx.x;

  float sp = 0.0f, st = 0.0f;
  for (int i = tid; i < NTOT; i += THREADS) {
    sp += rank[i];
    st += rank[NTOT + i];
  }
  sA[tid] = sp;
  sB[tid] = st;
  __syncthreads();
  for (int off = THREADS / 2; off > 0; off >>= 1) {
    if (tid < off) {
      sA[tid] += sA[tid + off];
      sB[tid] += sB[tid + off];
    }
    __syncthreads();
  }
  const float mp = sA[0] * (1.0f / NTOT);
  const float mt = sB[0] * (1.0f / NTOT);
  __syncthreads();

  float num = 0.0f, dp = 0.0f, dt = 0.0f;
  for (int i = tid; i < NTOT; i += THREADS) {
    float p = rank[i] - mp;
    float t = rank[NTOT + i] - mt;
    num = __builtin_fmaf(p, t, num);
    dp = __builtin_fmaf(p, p, dp);
    dt = __builtin_fmaf(t, t, dt);
  }
  sA[tid] = num;
  sB[tid] = dp;
  sC[tid] = dt;
  __syncthreads();
  for (int off = THREADS / 2; off > 0; off >>= 1) {
    if (tid < off) {
      sA[tid] += sA[tid + off];
      sB[tid] += sB[tid + off];
      sC[tid] += sC[tid + off];
    }
    __syncthreads();
  }
  if (tid == 0) {
    out[0] = -(sA[0] / (sqrtf(sB[0] * sC[0]) + 1e-8f));
  }
}

extern "C" void kernel_launch(void* const* d_in, const int* in_sizes, int n_in,
                              void* d_out, int out_size, void* d_ws,
                              size_t ws_size, hipStream_t stream) {
  (void)in_sizes;
  (void)n_in;
  (void)out_size;
  (void)ws_size;
  const float* pred = (const float*)d_in[0];
  const float* targ = (const float*)d_in[1];
  float* rank = (float*)d_ws;  // 2*NTOT floats = 64 KB of scratch
  float* out = (float*)d_out;

  zero_ws_kernel<<<(2 * NTOT + THREADS - 1) / THREADS, THREADS, 0, stream>>>(
      rank, 2 * NTOT);
  soft_rank_partial<<<dim3(NTOT / ROWS_PER_BLOCK, JSPLIT, 2), THREADS, 0,
                      stream>>>(pred, targ, rank);
  ic_finalize<<<1, THREADS, 0, stream>>>(rank, out);
}